// VQC_81767587381317
// MI455X (gfx1250) — compile-verified
//
#include <hip/hip_runtime.h>

#define NQ 12
#define NSTATE 4096
#define NL 4
#define NC 10
#define BATCH 512

typedef float v2f __attribute__((ext_vector_type(2)));
typedef float v8f __attribute__((ext_vector_type(8)));

// Flat state index for row r (non-group bits) and group-local index t.
// Qubit q lives at bit position (11 - q). Group G covers qubits 4G..4G+3,
// with qubit 4G at t-bit 3 (most significant of the group).
template <int G>
__device__ __forceinline__ int sidxT(int r, int t) {
  if constexpr (G == 0) return (t << 8) | r;                             // bits 11..8
  else if constexpr (G == 1) return ((r & 0xF0) << 4) | (t << 4) | (r & 15);  // bits 7..4
  else return (r << 4) | t;                                              // bits 3..0
}

// Composed CNOT-ring permutation: CNOT(q,(q+1)%12) applied for q=0..11 in order.
__device__ __forceinline__ int cnot_perm(int i) {
  int b = i;
#pragma unroll
  for (int q = 0; q < 12; ++q) {
    int cpos = 11 - q;
    int tpos = 11 - ((q + 1) % 12);
    b ^= ((b >> cpos) & 1) << tpos;
  }
  return b;
}

// state = state * E^T, E real 16x16 (applied to Re and Im independently).
template <int G>
__device__ __forceinline__ void apply_real(float* sRe, float* sIm, const v2f br[4],
                                           int wid, int n16, int k0, int hi8) {
#pragma unroll
  for (int tt = 0; tt < 2; ++tt) {
    int tile = (wid * 2 + tt) * 16;
    v2f ar[4], ai[4];
#pragma unroll
    for (int cc = 0; cc < 4; ++cc) {
      int k = 4 * cc + k0;
      int r = tile + n16;
      int i0 = sidxT<G>(r, k), i1 = sidxT<G>(r, k + 1);
      ar[cc].x = sRe[i0]; ar[cc].y = sRe[i1];
      ai[cc].x = sIm[i0]; ai[cc].y = sIm[i1];
    }
    v8f dr = {0, 0, 0, 0, 0, 0, 0, 0};
    v8f di = {0, 0, 0, 0, 0, 0, 0, 0};
#pragma unroll
    for (int cc = 0; cc < 4; ++cc) {
      dr = __builtin_amdgcn_wmma_f32_16x16x4_f32(false, ar[cc], false, br[cc],
                                                 (short)0, dr, false, false);
      di = __builtin_amdgcn_wmma_f32_16x16x4_f32(false, ai[cc], false, br[cc],
                                                 (short)0, di, false, false);
    }
#pragma unroll
    for (int v = 0; v < 8; ++v) {
      int r = tile + v + hi8;
      int o = sidxT<G>(r, n16);
      sRe[o] = dr[v];
      sIm[o] = di[v];
    }
  }
}

// state = state * U^T, U complex 16x16 stored row-major in (Ur, Ui).
template <int G>
__device__ __forceinline__ void apply_cplx(float* sRe, float* sIm,
                                           const float* Ur, const float* Ui,
                                           int wid, int n16, int k0, int hi8) {
  v2f br[4], bi[4], bin[4];
#pragma unroll
  for (int cc = 0; cc < 4; ++cc) {
    int k = 4 * cc + k0;
    br[cc].x = Ur[n16 * 16 + k];  br[cc].y = Ur[n16 * 16 + k + 1];
    bi[cc].x = Ui[n16 * 16 + k];  bi[cc].y = Ui[n16 * 16 + k + 1];
    bin[cc].x = -bi[cc].x;        bin[cc].y = -bi[cc].y;
  }
#pragma unroll
  for (int tt = 0; tt < 2; ++tt) {
    int tile = (wid * 2 + tt) * 16;
    v2f ar[4], ai[4];
#pragma unroll
    for (int cc = 0; cc < 4; ++cc) {
      int k = 4 * cc + k0;
      int r = tile + n16;
      int i0 = sidxT<G>(r, k), i1 = sidxT<G>(r, k + 1);
      ar[cc].x = sRe[i0]; ar[cc].y = sRe[i1];
      ai[cc].x = sIm[i0]; ai[cc].y = sIm[i1];
    }
    v8f dr = {0, 0, 0, 0, 0, 0, 0, 0};
    v8f di = {0, 0, 0, 0, 0, 0, 0, 0};
#pragma unroll
    for (int cc = 0; cc < 4; ++cc) {
      // out_re = S_re*Bre + S_im*(-Bim) ; out_im = S_re*Bim + S_im*Bre
      dr = __builtin_amdgcn_wmma_f32_16x16x4_f32(false, ar[cc], false, br[cc],
                                                 (short)0, dr, false, false);
      dr = __builtin_amdgcn_wmma_f32_16x16x4_f32(false, ai[cc], false, bin[cc],
                                                 (short)0, dr, false, false);
      di = __builtin_amdgcn_wmma_f32_16x16x4_f32(false, ar[cc], false, bi[cc],
                                                 (short)0, di, false, false);
      di = __builtin_amdgcn_wmma_f32_16x16x4_f32(false, ai[cc], false, br[cc],
                                                 (short)0, di, false, false);
    }
#pragma unroll
    for (int v = 0; v < 8; ++v) {
      int r = tile + v + hi8;
      int o = sidxT<G>(r, n16);
      sRe[o] = dr[v];
      sIm[o] = di[v];
    }
  }
}

__global__ __launch_bounds__(256) void vqc_kernel(const float* __restrict__ x,
                                                  const float* __restrict__ w,
                                                  const float* __restrict__ bias,
                                                  float* __restrict__ probs_out) {
  __shared__ float sRe[NSTATE];
  __shared__ float sIm[NSTATE];
  __shared__ float Eg[3 * 256];       // fused real RY matrices (layer-invariant)
  __shared__ float Rre[12 * 256];     // fused complex Rot matrices, mat = l*3+g
  __shared__ float Rim[12 * 256];
  __shared__ float ryc[NQ], rys[NQ];  // per-qubit RY cos/sin
  __shared__ float rotRe[48 * 4];     // per (l,q) Rot 2x2 entries
  __shared__ float rotIm[48 * 4];
  __shared__ float zw[8 * NC];

  const int tid  = threadIdx.x;
  const int s    = blockIdx.x;
  const int lane = tid & 31;
  const int wid  = tid >> 5;               // 8 waves
  const int n16  = lane & 15;              // N column / M row within tile
  const int k0   = ((lane >> 4) & 1) << 1; // K base: lanes 0-15 -> 0, 16-31 -> 2
  const int hi8  = ((lane >> 4) & 1) << 3; // D rows: lanes 16-31 hold M+8

  // ---- stage 1: per-qubit gate entries (all trig happens here, once) ----
  if (tid < NQ) {
    float h = 0.5f * x[s * NQ + tid];
    ryc[tid] = cosf(h);
    rys[tid] = sinf(h);
  }
  if (tid < NL * NQ) {  // tid = l*12+q
    const float* wp = &w[tid * 3];
    float phi = wp[0], th = wp[1], om = wp[2];
    float c = cosf(0.5f * th), sn = sinf(0.5f * th);
    float a = 0.5f * (phi + om), b = 0.5f * (phi - om);
    float ca = cosf(a), sa = sinf(a), cb = cosf(b), sb = sinf(b);
    float* gr = &rotRe[tid * 4];
    float* gi = &rotIm[tid * 4];
    gr[0] = c * ca;    gi[0] = -c * sa;   // (0,0)
    gr[1] = -sn * cb;  gi[1] = -sn * sb;  // (0,1)
    gr[2] = sn * cb;   gi[2] = -sn * sb;  // (1,0)
    gr[3] = c * ca;    gi[3] = c * sa;    // (1,1)
  }
  // init |0...0>
#pragma unroll
  for (int e = 0; e < 16; ++e) {
    sRe[e * 256 + tid] = 0.f;
    sIm[e * 256 + tid] = 0.f;
  }
  __syncthreads();
  if (tid == 0) sRe[0] = 1.f;

  // ---- stage 2: build all fused 16x16 matrices (products only, no trig) ----
#pragma unroll
  for (int m = 0; m < 3; ++m) {  // 3*256 embedding entries
    int idx = m * 256 + tid;
    int g = idx >> 8, k = idx & 255;
    int tp = k >> 4, t = k & 15;
    float prod = 1.f;
#pragma unroll
    for (int j = 0; j < 4; ++j) {
      int q = 4 * g + j;
      float c = ryc[q], sn = rys[q];
      int i2 = (tp >> (3 - j)) & 1;
      int j2 = (t >> (3 - j)) & 1;
      prod *= i2 ? (j2 ? c : sn) : (j2 ? -sn : c);
    }
    Eg[idx] = prod;
  }
#pragma unroll
  for (int m = 0; m < 12; ++m) {  // 12*256 Rot entries, mat = l*3+g
    int idx = m * 256 + tid;
    int mat = idx >> 8, k = idx & 255;
    int l = mat / 3, g = mat % 3;
    int tp = k >> 4, t = k & 15;
    float pr = 1.f, pi = 0.f;
#pragma unroll
    for (int j = 0; j < 4; ++j) {
      int lq = l * NQ + 4 * g + j;
      int i2 = (tp >> (3 - j)) & 1;
      int j2 = (t >> (3 - j)) & 1;
      float er = rotRe[lq * 4 + i2 * 2 + j2];
      float ei = rotIm[lq * 4 + i2 * 2 + j2];
      float nr = pr * er - pi * ei;
      float ni = pr * ei + pi * er;
      pr = nr; pi = ni;
    }
    Rre[idx] = pr;
    Rim[idx] = pi;
  }
  __syncthreads();

  // embedding B-operands are layer-invariant: load once, keep in VGPRs
  v2f brE[3][4];
#pragma unroll
  for (int g = 0; g < 3; ++g) {
#pragma unroll
    for (int cc = 0; cc < 4; ++cc) {
      int k = 4 * cc + k0;
      brE[g][cc].x = Eg[g * 256 + n16 * 16 + k];
      brE[g][cc].y = Eg[g * 256 + n16 * 16 + k + 1];
    }
  }

  // ---- circuit: 4 layers of pure WMMA GEMMs + basis permutation ----
  for (int l = 0; l < NL; ++l) {
    apply_real<0>(sRe, sIm, brE[0], wid, n16, k0, hi8); __syncthreads();
    apply_real<1>(sRe, sIm, brE[1], wid, n16, k0, hi8); __syncthreads();
    apply_real<2>(sRe, sIm, brE[2], wid, n16, k0, hi8); __syncthreads();

    const float* R0r = &Rre[(l * 3 + 0) * 256]; const float* R0i = &Rim[(l * 3 + 0) * 256];
    const float* R1r = &Rre[(l * 3 + 1) * 256]; const float* R1i = &Rim[(l * 3 + 1) * 256];
    const float* R2r = &Rre[(l * 3 + 2) * 256]; const float* R2i = &Rim[(l * 3 + 2) * 256];
    apply_cplx<0>(sRe, sIm, R0r, R0i, wid, n16, k0, hi8); __syncthreads();
    apply_cplx<1>(sRe, sIm, R1r, R1i, wid, n16, k0, hi8); __syncthreads();
    apply_cplx<2>(sRe, sIm, R2r, R2i, wid, n16, k0, hi8); __syncthreads();

    // CNOT ring: composed basis permutation, staged via registers
    float vr[16], vi[16];
    int dst[16];
#pragma unroll
    for (int e = 0; e < 16; ++e) {
      int i = e * 256 + tid;
      vr[e] = sRe[i];
      vi[e] = sIm[i];
      dst[e] = cnot_perm(i);
    }
    __syncthreads();
#pragma unroll
    for (int e = 0; e < 16; ++e) {
      sRe[dst[e]] = vr[e];
      sIm[dst[e]] = vi[e];
    }
    __syncthreads();
  }

  // ---- <Z_c> for c=0..9, then softmax ----
  float z[NC];
#pragma unroll
  for (int c = 0; c < NC; ++c) z[c] = 0.f;
#pragma unroll
  for (int e = 0; e < 16; ++e) {
    int i = e * 256 + tid;
    float p = sRe[i] * sRe[i] + sIm[i] * sIm[i];
#pragma unroll
    for (int c = 0; c < NC; ++c) z[c] += ((i >> (11 - c)) & 1) ? -p : p;
  }
#pragma unroll
  for (int c = 0; c < NC; ++c) {
#pragma unroll
    for (int off = 16; off > 0; off >>= 1) z[c] += __shfl_xor(z[c], off);
  }
  if (lane == 0) {
#pragma unroll
    for (int c = 0; c < NC; ++c) zw[wid * NC + c] = z[c];
  }
  __syncthreads();
  if (tid == 0) {
    float ez[NC];
    float mx = -1e30f;
    for (int c = 0; c < NC; ++c) {
      float acc = 0.f;
      for (int wv = 0; wv < 8; ++wv) acc += zw[wv * NC + c];
      ez[c] = acc + bias[c];
      mx = fmaxf(mx, ez[c]);
    }
    float ssum = 0.f;
    for (int c = 0; c < NC; ++c) { ez[c] = expf(ez[c] - mx); ssum += ez[c]; }
    float inv = 1.f / ssum;
    for (int c = 0; c < NC; ++c) probs_out[s * NC + c] = ez[c] * inv;
  }
}

// BatchNorm1d over the batch axis (training-mode biased stats), one block per class.
__global__ __launch_bounds__(512) void bn_kernel(const float* __restrict__ P,
                                                 const float* __restrict__ gamma,
                                                 const float* __restrict__ beta,
                                                 float* __restrict__ out) {
  __shared__ float s1[512];
  __shared__ float s2[512];
  int c = blockIdx.x;
  int t = threadIdx.x;
  float p = P[t * NC + c];
  s1[t] = p;
  s2[t] = p * p;
  __syncthreads();
  for (int off = 256; off > 0; off >>= 1) {
    if (t < off) {
      s1[t] += s1[t + off];
      s2[t] += s2[t + off];
    }
    __syncthreads();
  }
  float mu = s1[0] * (1.f / 512.f);
  float var = s2[0] * (1.f / 512.f) - mu * mu;
  out[t * NC + c] = (p - mu) * rsqrtf(var + 1e-5f) * gamma[c] + beta[c];
}

extern "C" void kernel_launch(void* const* d_in, const int* in_sizes, int n_in,
                              void* d_out, int out_size, void* d_ws, size_t ws_size,
                              hipStream_t stream) {
  const float* x     = (const float*)d_in[0];
  const float* w     = (const float*)d_in[1];
  const float* bias  = (const float*)d_in[2];
  const float* gamma = (const float*)d_in[3];
  const float* beta  = (const float*)d_in[4];
  float* out = (float*)d_out;
  float* P   = (float*)d_ws;  // 512*10 softmax probs

  vqc_kernel<<<BATCH, 256, 0, stream>>>(x, w, bias, P);
  bn_kernel<<<NC, 512, 0, stream>>>(P, gamma, beta, out);
}